// MinMaxMeanPooling_5282809774456
// MI455X (gfx1250) — compile-verified
//
#include <hip/hip_runtime.h>
#include <stdint.h>

typedef __attribute__((ext_vector_type(2))) float v2f;
typedef __attribute__((ext_vector_type(8))) float v8f;

#define RPW 1024  // rows per block (one contiguous range per block)

__device__ __forceinline__ int imin(int a, int b) { return a < b ? a : b; }

// Raw v_min_num/v_max_num accumulate: avoids the backend's per-use input
// canonicalization that fminf/fmaxf (llvm.minnum) trigger on gfx1250.
__device__ __forceinline__ void vmin_acc(float& acc, float v) {
  asm("v_min_num_f32 %0, %0, %1" : "+v"(acc) : "v"(v));
}
__device__ __forceinline__ void vmax_acc(float& acc, float v) {
  asm("v_max_num_f32 %0, %0, %1" : "+v"(acc) : "v"(v));
}

// Native CDNA5 float min/max atomics (no-return form, device scope).
__device__ __forceinline__ void atomic_min_f32(float* p, float v) {
  asm volatile("global_atomic_min_num_f32 %0, %1, off scope:SCOPE_DEV" :: "v"(p), "v"(v) : "memory");
}
__device__ __forceinline__ void atomic_max_f32(float* p, float v) {
  asm volatile("global_atomic_max_num_f32 %0, %1, off scope:SCOPE_DEV" :: "v"(p), "v"(v) : "memory");
}

__global__ void init_ws_kernel(float* __restrict__ wmin, float* __restrict__ wmax,
                               float* __restrict__ wsum, float* __restrict__ wcnt,
                               int GH, int G) {
  int i = blockIdx.x * blockDim.x + threadIdx.x;
  if (i < GH) {
    wmin[i] = __builtin_inff();
    wmax[i] = -__builtin_inff();
    wsum[i] = 0.0f;
  }
  if (i < G) wcnt[i] = 0.0f;
}

// HH != 0: compile-time feature width (hot instantiation, immediate load offsets).
// HH == 0: runtime fallback using Hrt.
template <int HH>
__global__ void pool_kernel_t(const float* __restrict__ x, const int* __restrict__ batch,
                              float* __restrict__ wmin, float* __restrict__ wmax,
                              float* __restrict__ wsum, float* __restrict__ wcnt,
                              int N, int Hrt) {
  const int H = HH ? HH : Hrt;
  const int lane  = threadIdx.x & 31;
  const int wid   = threadIdx.x >> 5;   // feature chunk within block
  const int fbase = wid * 16;
  const int r0 = blockIdx.x * RPW;
  const int r1 = imin(r0 + RPW, N);
  if (r0 >= N) return;

  const int ln = lane & 15;
  const int sr = (lane < 16) ? 0 : 2;   // row slot carried by B vgpr0 in this lane half
  const float INF = __builtin_inff();

  // A = ones(16x4): D[m][n] = colsum(B)[n] + C[m][n]; all D rows identical.
  v2f ones; ones.x = 1.0f; ones.y = 1.0f;

  const unsigned coloff = (unsigned)(fbase + ln); // this lane's feature column
  const char* xb = (const char*)x;

  int r = r0;
  while (r < r1) {
    const int rs = r;
    const int cur = __builtin_amdgcn_readfirstlane(batch[r]);

    // Sorted batch: binary-search the end of this run within [r+1, r1).
    int lo = r + 1, hi = r1;
    while (lo < hi) {
      int mid = (lo + hi) >> 1;
      int v = __builtin_amdgcn_readfirstlane(batch[mid]);
      if (v == cur) lo = mid + 1; else hi = mid;
    }
    const int e = lo;                    // run is [rs, e), all rows == cur

    v8f csA = {}, csB = {};
    float mnA0 = INF, mnA1 = INF, mxA0 = -INF, mxA1 = -INF;
    float mnB0 = INF, mnB1 = INF, mxB0 = -INF, mxB1 = -INF;

    // Per-lane BYTE offset induction (N*H*4 < 2^32); constant byte offsets
    // (4H, 16H, 20H with H compile-time) fold into the load IOFFSET field.
    unsigned offB = (coloff + (unsigned)(r + sr) * (unsigned)H) * 4u;

    // Hot loop: 8 rows/iter, no segment checks, dual WMMA accumulators.
    for (; r + 8 <= e; r += 8, offB += 32u * (unsigned)H) {
      float a0 = *(const float*)(xb + offB);
      float a1 = *(const float*)(xb + offB + 4u  * (unsigned)H);
      float b0 = *(const float*)(xb + offB + 16u * (unsigned)H);
      float b1 = *(const float*)(xb + offB + 20u * (unsigned)H);
      v2f BA; BA.x = a0; BA.y = a1;
      v2f BB; BB.x = b0; BB.y = b1;
      csA = __builtin_amdgcn_wmma_f32_16x16x4_f32(false, ones, false, BA, (short)0, csA, false, false);
      csB = __builtin_amdgcn_wmma_f32_16x16x4_f32(false, ones, false, BB, (short)0, csB, false, false);
      vmin_acc(mnA0, a0); vmin_acc(mnA1, a1);
      vmax_acc(mxA0, a0); vmax_acc(mxA1, a1);
      vmin_acc(mnB0, b0); vmin_acc(mnB1, b1);
      vmax_acc(mxB0, b0); vmax_acc(mxB1, b1);
    }
    // At most one remaining full 4-row group.
    if (r + 4 <= e) {
      float a0 = *(const float*)(xb + offB);
      float a1 = *(const float*)(xb + offB + 4u * (unsigned)H);
      v2f BA; BA.x = a0; BA.y = a1;
      csA = __builtin_amdgcn_wmma_f32_16x16x4_f32(false, ones, false, BA, (short)0, csA, false, false);
      vmin_acc(mnA0, a0); vmin_acc(mnA1, a1);
      vmax_acc(mxA0, a0); vmax_acc(mxA1, a1);
      r += 4;
      offB += 16u * (unsigned)H;
    }
    // Tail (1..3 rows) — all inside this run; mask unused slots to identities.
    // Clamp addresses (rows r+sr(+1) may exceed N-1 for masked slots).
    const int t = e - r;
    if (t > 0) {
      float a0 = *(const float*)(xb + (coloff + (unsigned)imin(r + sr,     N - 1) * (unsigned)H) * 4u);
      float a1 = *(const float*)(xb + (coloff + (unsigned)imin(r + sr + 1, N - 1) * (unsigned)H) * 4u);
      bool h0 = (sr < t), h1 = (sr + 1 < t);
      v2f BA; BA.x = h0 ? a0 : 0.0f; BA.y = h1 ? a1 : 0.0f;
      csA = __builtin_amdgcn_wmma_f32_16x16x4_f32(false, ones, false, BA, (short)0, csA, false, false);
      vmin_acc(mnA0, h0 ? a0 :  INF); vmin_acc(mnA1, h1 ? a1 :  INF);
      vmax_acc(mxA0, h0 ? a0 : -INF); vmax_acc(mxA1, h1 ? a1 : -INF);
      r = e;
    }

    // Flush this run. Column sum (replicated across all D rows) is element 0 of
    // each accumulator; lanes n and n+16 hold the same value -> add from lanes
    // 0..15 only. Min/max: both lane halves cover the same feature for different
    // row slots; both issue atomics (idempotent under min/max).
    {
      float s = csA[0] + csB[0];
      float pmn = mnA0, pmx = mxA0;
      vmin_acc(pmn, mnA1); vmin_acc(pmn, mnB0); vmin_acc(pmn, mnB1);
      vmax_acc(pmx, mxA1); vmax_acc(pmx, mxB0); vmax_acc(pmx, mxB1);
      size_t o = (size_t)cur * H + fbase + ln;
      atomic_min_f32(&wmin[o], pmn);
      atomic_max_f32(&wmax[o], pmx);
      if (lane < 16) atomicAdd(&wsum[o], s);
      if (wid == 0 && lane == 0) atomicAdd(&wcnt[cur], (float)(e - rs));
    }
  }
}

__global__ void finalize_kernel(const float* __restrict__ wmin, const float* __restrict__ wmax,
                                const float* __restrict__ wsum, const float* __restrict__ wcnt,
                                float* __restrict__ out, int G, int H) {
  int i = blockIdx.x * blockDim.x + threadIdx.x;
  if (i >= G * H) return;
  int g = i / H, h = i - g * H;
  float c = wcnt[g];
  bool nonempty = c > 0.0f;
  float mn = nonempty ? wmin[i] : 0.0f;
  float mx = nonempty ? wmax[i] : 0.0f;
  float mean = wsum[i] / fmaxf(c, 1.0f);
  size_t base = (size_t)g * (3 * H);
  out[base + h]         = mn;
  out[base + H + h]     = mx;
  out[base + 2 * H + h] = mean;
}

extern "C" void kernel_launch(void* const* d_in, const int* in_sizes, int n_in,
                              void* d_out, int out_size, void* d_ws, size_t ws_size,
                              hipStream_t stream) {
  const float* x     = (const float*)d_in[0];
  const int*   batch = (const int*)d_in[1];
  const int N = in_sizes[1];
  const int H = in_sizes[0] / N;          // 128
  const int G = out_size / (3 * H);       // 4096

  float* wmin = (float*)d_ws;
  float* wmax = wmin + (size_t)G * H;
  float* wsum = wmax + (size_t)G * H;
  float* wcnt = wsum + (size_t)G * H;
  const int GH = G * H;

  init_ws_kernel<<<(GH + 255) / 256, 256, 0, stream>>>(wmin, wmax, wsum, wcnt, GH, G);

  const int nRanges = (N + RPW - 1) / RPW;
  const int tpb = (H / 16) * 32;          // one wave per 16-feature chunk (256 for H=128)
  if (H == 128) {
    pool_kernel_t<128><<<nRanges, tpb, 0, stream>>>(x, batch, wmin, wmax, wsum, wcnt, N, H);
  } else {
    pool_kernel_t<0><<<nRanges, tpb, 0, stream>>>(x, batch, wmin, wmax, wsum, wcnt, N, H);
  }

  finalize_kernel<<<(GH + 255) / 256, 256, 0, stream>>>(wmin, wmax, wsum, wcnt,
                                                        (float*)d_out, G, H);
}